// LSTM_47751446397283
// MI455X (gfx1250) — compile-verified
//
#include <hip/hip_runtime.h>
#include <hip/hip_bf16.h>

typedef float v2f __attribute__((ext_vector_type(2)));
typedef float v8f __attribute__((ext_vector_type(8)));
typedef unsigned int v4u __attribute__((ext_vector_type(4)));
typedef int v4i __attribute__((ext_vector_type(4)));
typedef int v8i __attribute__((ext_vector_type(8)));

#if __has_builtin(__builtin_amdgcn_tensor_load_to_lds) && __has_builtin(__builtin_amdgcn_s_wait_tensorcnt)
#define USE_TDM 1
#else
#define USE_TDM 0
#endif

namespace {
constexpr int kB = 128;   // batch
constexpr int kT = 1024;  // encoder time steps
constexpr int kI = 512;   // input feature
constexpr int kH = 1024;  // hidden
constexpr int kO = 512;   // output feature
constexpr int kL = 128;   // decoder steps
constexpr int kG = 4 * kH;         // 4096 gate columns
constexpr int kKenc = kI + kH;     // 1536

// workspace layout (float offsets)
constexpr size_t WT_OFF   = 0;                           // Wcat^T  (kKenc x kG), k-major
constexpr size_t WT_SZ    = (size_t)kKenc * kG;
constexpr size_t WOUT_OFF = WT_OFF + WT_SZ;              // W_out^T (kH x kO), k-major
constexpr size_t WOUT_SZ  = (size_t)kH * kO;
constexpr size_t BIAS_OFF = WOUT_OFF + WOUT_SZ;          // b_ih + b_hh (kG)
constexpr size_t BIAS_SZ  = (size_t)kG;
constexpr size_t HC_SZ    = (size_t)kB * kH;
constexpr size_t H_OFF    = BIAS_OFF + BIAS_SZ;          // 2 buffers
constexpr size_t C_OFF    = H_OFF + 2 * HC_SZ;           // 2 buffers
}

__device__ __forceinline__ float fsig(float x) {
  return 1.0f / (1.0f + __expf(-x));
}
__device__ __forceinline__ float ftanh(float x) {
  float xc = fminf(fmaxf(x, -15.0f), 15.0f);
  float e  = __expf(-2.0f * xc);
  return (1.0f - e) / (1.0f + e);
}

#if USE_TDM
// TDM: async-load a 32x32 fp32 tile (rows strided by row_stride elements)
// into LDS at lds_off, padding 1 dword after every 32 dwords (row stride 33).
// D# group0/group1 packed per CDNA5 ISA ch.8 (2-D tile, groups 2/3 unused).
__device__ __forceinline__ void tdm_load_tile32x32(unsigned lds_off,
                                                   const float* gsrc,
                                                   unsigned row_stride) {
  const unsigned long long ga = (unsigned long long)gsrc;
  v4u g0;
  g0[0] = 1u;                                  // count=1, user mode
  g0[1] = lds_off;                             // lds_addr (bytes)
  g0[2] = (unsigned)(ga & 0xFFFFFFFFull);      // global_addr[31:0]
  g0[3] = (unsigned)((ga >> 32) & 0x1FFFFFFull) | (2u << 30);  // addr[56:32] | type=2
  v8i g1;
  g1[0] = (2 << 16) | (1 << 20) | (4 << 22);   // data_size=4B, pad_enable, interval=32dw (pad=1dw)
  g1[1] = 32 << 16;                            // tensor_dim0 = 32
  g1[2] = 32 << 16;                            // tensor_dim1 = 32 (dim0 hi = 0)
  g1[3] = 32 << 16;                            // tile_dim0 = 32 (dim1 hi = 0)
  g1[4] = 32;                                  // tile_dim1 = 32, tile_dim2 = 0
  g1[5] = (int)row_stride;                     // tensor_dim0_stride[31:0]
  g1[6] = 0;                                   // stride hi, dim1_stride lo
  g1[7] = 0;
  v4i z = {};
#if defined(__clang_major__) && (__clang_major__ >= 23)
  v8i z8 = {};
  __builtin_amdgcn_tensor_load_to_lds(g0, g1, z, z, z8, 0);
#else
  __builtin_amdgcn_tensor_load_to_lds(g0, g1, z, z, 0);
#endif
}
#endif

// ---------------------------------------------------------------------------
// One-shot prep: transpose/concat weights k-major, fuse biases, zero h0/c0.
// ---------------------------------------------------------------------------
__global__ void lstm_prep(const float* __restrict__ W_ih,
                          const float* __restrict__ W_hh,
                          const float* __restrict__ b_ih,
                          const float* __restrict__ b_hh,
                          const float* __restrict__ W_out,
                          float* __restrict__ ws) {
  float* Wt     = ws + WT_OFF;
  float* Wout_t = ws + WOUT_OFF;
  float* bias   = ws + BIAS_OFF;
  float* h0     = ws + H_OFF;
  float* c0     = ws + C_OFF;
  const size_t total = WT_SZ + WOUT_SZ + BIAS_SZ + 2 * HC_SZ;
  const size_t stride = (size_t)gridDim.x * blockDim.x;
  for (size_t i = (size_t)blockIdx.x * blockDim.x + threadIdx.x; i < total; i += stride) {
    if (i < WT_SZ) {
      size_t k = i / kG, n = i % kG;
      Wt[i] = (k < (size_t)kI) ? W_ih[n * kI + k] : W_hh[n * kH + (k - kI)];
    } else if (i < WT_SZ + WOUT_SZ) {
      size_t j = i - WT_SZ, h = j / kO, o = j % kO;
      Wout_t[j] = W_out[o * kH + h];
    } else if (i < WT_SZ + WOUT_SZ + BIAS_SZ) {
      size_t j = i - WT_SZ - WOUT_SZ;
      bias[j] = b_ih[j] + b_hh[j];
    } else if (i < WT_SZ + WOUT_SZ + BIAS_SZ + HC_SZ) {
      h0[i - (WT_SZ + WOUT_SZ + BIAS_SZ)] = 0.0f;
    } else {
      c0[i - (WT_SZ + WOUT_SZ + BIAS_SZ + HC_SZ)] = 0.0f;
    }
  }
}

// ---------------------------------------------------------------------------
// One LSTM step.  Workgroup = 2 wave32, tile = 32 (batch) x 32 (hidden cols).
// Wave w owns N-subtile w; it computes BOTH 16-row M-subtiles for all four
// gates (8 accumulators) so every B fragment feeds two WMMAs.
// A-tiles are TDM-DMA'd into double-buffered LDS (async, TENSORcnt).
// ---------------------------------------------------------------------------
template <bool ENC>
__global__ __launch_bounds__(64)
void lstm_step(const float* __restrict__ x, int t,
               const float* __restrict__ Wt,      // k-major (K x 4096), pre-offset for DEC
               const float* __restrict__ biasg,
               const int* __restrict__ lengths,   // int32
               const float* __restrict__ h_in,
               const float* __restrict__ c_in,
               float* __restrict__ h_out,
               float* __restrict__ c_out) {
  constexpr int KTOT = ENC ? kKenc : kH;
  constexpr int NCH  = KTOT / 32;
  __shared__ float As[2][32][33];

  const int tid  = threadIdx.x;
  const int lane = tid & 31;
  const int wid  = tid >> 5;                       // = N subtile
  const int m0   = blockIdx.y * 32;
  const int ncol = blockIdx.x * 32 + wid * 16 + (lane & 15);

  v8f acc[4][2] = {};   // [gate][m-subtile]

#if USE_TDM
  auto issue = [&](int c) {
    const int kc = c * 32;
    const float* gsrc;
    unsigned rs;
    if (ENC && kc < kI) {
      gsrc = x + ((size_t)m0 * kT + t) * kI + kc;
      rs   = (unsigned)(kT * kI);
    } else {
      gsrc = h_in + (size_t)m0 * kH + (kc - (ENC ? kI : 0));
      rs   = (unsigned)kH;
    }
    tdm_load_tile32x32((unsigned)(unsigned long long)&As[c & 1][0][0], gsrc, rs);
  };
  if (wid == 0) issue(0);
#endif

  for (int c = 0; c < NCH; ++c) {
    const int kc = c * 32;
#if USE_TDM
    if (wid == 0) {
      if (c + 1 < NCH) {
        issue(c + 1);                              // overlap next DMA with compute
        __builtin_amdgcn_s_wait_tensorcnt(1);      // retire chunk c only
      } else {
        __builtin_amdgcn_s_wait_tensorcnt(0);
      }
    }
    __syncthreads();
    const int buf = c & 1;
#else
    __syncthreads();
    {  // vector staging fallback: 64 threads x 16 floats
      const int ar = tid >> 1, ac = (tid & 1) * 16;
      const float* src;
      if (ENC && kc < kI)
        src = x + ((size_t)(m0 + ar) * kT + t) * kI + (kc + ac);
      else
        src = h_in + (size_t)(m0 + ar) * kH + (kc + ac - (ENC ? kI : 0));
#pragma unroll
      for (int j = 0; j < 16; ++j) As[0][ar][ac + j] = src[j];
    }
    __syncthreads();
    const int buf = 0;
#endif

    if (kc + 32 < KTOT)
      __builtin_prefetch(Wt + (size_t)(kc + 32 + ((lane >> 4) << 1)) * kG + ncol, 0, 1);

#pragma unroll
    for (int kk = 0; kk < 32; kk += 4) {
      const int kb = kk + ((lane >> 4) << 1);      // this lane-half's K pair
      v2f a0, a1;
      a0.x = As[buf][lane & 15][kb];
      a0.y = As[buf][lane & 15][kb + 1];
      a1.x = As[buf][16 + (lane & 15)][kb];
      a1.y = As[buf][16 + (lane & 15)][kb + 1];
      const float* wp = Wt + (size_t)(kc + kb) * kG + ncol;
#pragma unroll
      for (int g = 0; g < 4; ++g) {
        v2f b;
        b.x = wp[g * kH];
        b.y = wp[g * kH + kG];
        acc[g][0] = __builtin_amdgcn_wmma_f32_16x16x4_f32(
            false, a0, false, b, (short)0, acc[g][0], false, false);
        acc[g][1] = __builtin_amdgcn_wmma_f32_16x16x4_f32(
            false, a1, false, b, (short)0, acc[g][1], false, false);
      }
    }
    __syncthreads();
  }

  // fused cell epilogue
  const float bi = biasg[ncol];
  const float bf = biasg[ncol + kH];
  const float bg = biasg[ncol + 2 * kH];
  const float bo = biasg[ncol + 3 * kH];
  const int mhalf = (lane >> 4) * 8;               // C/D: lanes 16-31 hold M = v+8
#pragma unroll
  for (int mi = 0; mi < 2; ++mi) {
#pragma unroll
    for (int v = 0; v < 8; ++v) {
      const int gm = m0 + mi * 16 + v + mhalf;
      const size_t idx = (size_t)gm * kH + ncol;
      const float gi = acc[0][mi][v] + bi;
      const float gf = acc[1][mi][v] + bf;
      const float gg = acc[2][mi][v] + bg;
      const float go = acc[3][mi][v] + bo;
      const float cprev = c_in[idx];
      float ct = fsig(gf) * cprev + fsig(gi) * ftanh(gg);
      float ht = fsig(go) * ftanh(ct);
      if (ENC) {
        if (t >= lengths[gm]) { ct = cprev; ht = h_in[idx]; }
      }
      h_out[idx] = ht;
      c_out[idx] = ct;
    }
  }
}

// ---------------------------------------------------------------------------
// Output projection for decoder step l: out[:, l, :] = h @ W_out^T + b_out
// Same structure: 2 waves, wave owns 32x16, B fragment reused across M.
// ---------------------------------------------------------------------------
__global__ __launch_bounds__(64)
void out_proj(const float* __restrict__ h,        // (kB x kH)
              const float* __restrict__ Wout_t,   // k-major (kH x kO)
              const float* __restrict__ b_out,
              float* __restrict__ out, int l) {
  constexpr int NCH = kH / 32;
  __shared__ float As[2][32][33];
  const int tid  = threadIdx.x;
  const int lane = tid & 31;
  const int wid  = tid >> 5;
  const int m0   = blockIdx.y * 32;
  const int ncol = blockIdx.x * 32 + wid * 16 + (lane & 15);

  v8f acc[2] = {};

#if USE_TDM
  auto issue = [&](int c) {
    tdm_load_tile32x32((unsigned)(unsigned long long)&As[c & 1][0][0],
                       h + (size_t)m0 * kH + c * 32, (unsigned)kH);
  };
  if (wid == 0) issue(0);
#endif

  for (int c = 0; c < NCH; ++c) {
    const int kc = c * 32;
#if USE_TDM
    if (wid == 0) {
      if (c + 1 < NCH) {
        issue(c + 1);
        __builtin_amdgcn_s_wait_tensorcnt(1);
      } else {
        __builtin_amdgcn_s_wait_tensorcnt(0);
      }
    }
    __syncthreads();
    const int buf = c & 1;
#else
    __syncthreads();
    {
      const int ar = tid >> 1, ac = (tid & 1) * 16;
      const float* src = h + (size_t)(m0 + ar) * kH + (kc + ac);
#pragma unroll
      for (int j = 0; j < 16; ++j) As[0][ar][ac + j] = src[j];
    }
    __syncthreads();
    const int buf = 0;
#endif

#pragma unroll
    for (int kk = 0; kk < 32; kk += 4) {
      const int kb = kk + ((lane >> 4) << 1);
      v2f a0, a1;
      a0.x = As[buf][lane & 15][kb];
      a0.y = As[buf][lane & 15][kb + 1];
      a1.x = As[buf][16 + (lane & 15)][kb];
      a1.y = As[buf][16 + (lane & 15)][kb + 1];
      v2f b;
      b.x = Wout_t[(size_t)(kc + kb) * kO + ncol];
      b.y = Wout_t[(size_t)(kc + kb + 1) * kO + ncol];
      acc[0] = __builtin_amdgcn_wmma_f32_16x16x4_f32(
          false, a0, false, b, (short)0, acc[0], false, false);
      acc[1] = __builtin_amdgcn_wmma_f32_16x16x4_f32(
          false, a1, false, b, (short)0, acc[1], false, false);
    }
    __syncthreads();
  }

  const float bb = b_out[ncol];
  const int mhalf = (lane >> 4) * 8;
#pragma unroll
  for (int mi = 0; mi < 2; ++mi) {
#pragma unroll
    for (int v = 0; v < 8; ++v) {
      const int gm = m0 + mi * 16 + v + mhalf;
      out[((size_t)gm * kL + l) * kO + ncol] = acc[mi][v] + bb;
    }
  }
}

// ---------------------------------------------------------------------------
extern "C" void kernel_launch(void* const* d_in, const int* in_sizes, int n_in,
                              void* d_out, int out_size, void* d_ws, size_t ws_size,
                              hipStream_t stream) {
  const float* x       = (const float*)d_in[0];
  const int*   lengths = (const int*)d_in[1];
  const float* W_ih  = (const float*)d_in[3];
  const float* W_hh  = (const float*)d_in[4];
  const float* b_ih  = (const float*)d_in[5];
  const float* b_hh  = (const float*)d_in[6];
  const float* W_out = (const float*)d_in[7];
  const float* b_out = (const float*)d_in[8];
  float* out = (float*)d_out;
  float* ws  = (float*)d_ws;

  float* Wt     = ws + WT_OFF;
  float* Wout_t = ws + WOUT_OFF;
  float* bias   = ws + BIAS_OFF;
  float* hbuf   = ws + H_OFF;
  float* cbuf   = ws + C_OFF;

  lstm_prep<<<2048, 256, 0, stream>>>(W_ih, W_hh, b_ih, b_hh, W_out, ws);

  const dim3 sgrid(kH / 32, kB / 32);  // 32 x 4 = 128 workgroups
  for (int t = 0; t < kT; ++t) {
    const float* hi = hbuf + (size_t)(t & 1) * HC_SZ;
    const float* ci = cbuf + (size_t)(t & 1) * HC_SZ;
    float* ho = hbuf + (size_t)((t + 1) & 1) * HC_SZ;
    float* co = cbuf + (size_t)((t + 1) & 1) * HC_SZ;
    lstm_step<true><<<sgrid, 64, 0, stream>>>(x, t, Wt, bias, lengths, hi, ci, ho, co);
  }

  const dim3 pgrid(kO / 32, kB / 32);  // 16 x 4
  for (int l = 0; l < kL; ++l) {
    const float* hi = hbuf + (size_t)(l & 1) * HC_SZ;
    const float* ci = cbuf + (size_t)(l & 1) * HC_SZ;
    float* ho = hbuf + (size_t)((l + 1) & 1) * HC_SZ;
    float* co = cbuf + (size_t)((l + 1) & 1) * HC_SZ;
    lstm_step<false><<<sgrid, 64, 0, stream>>>(nullptr, 0, Wt + (size_t)kI * kG,
                                               bias, nullptr, hi, ci, ho, co);
    out_proj<<<pgrid, 64, 0, stream>>>(ho, Wout_t, b_out, out, l);
  }
  (void)in_sizes; (void)n_in; (void)out_size; (void)ws_size;
}